// AttentionDecoderRNN_56478819942575
// MI455X (gfx1250) — compile-verified
//
#include <hip/hip_runtime.h>
#include <math.h>

#define H 1024
#define V 50257
#define S 4096
#define NCH 16

typedef float v2f __attribute__((ext_vector_type(2)));
typedef float v8f __attribute__((ext_vector_type(8)));

// x = [embedding[word], last_context]  (2H)
__global__ void k_gather_x(const int* __restrict__ word, const float* __restrict__ emb,
                           const float* __restrict__ lastctx, float* __restrict__ x) {
    int i = blockIdx.x * blockDim.x + threadIdx.x;
    if (i < 2 * H) {
        x[i] = (i < H) ? emb[(size_t)word[0] * H + i] : lastctx[i - H];
    }
}

// out[row] = dot(W[row, 0:K], x) + bias[row] + *scalarAdd
// One wave per 16-row tile; K reduced 4 at a time via V_WMMA_F32_16X16X4_F32
// with B broadcast across all 16 columns (batch-1 GEMV, bandwidth-bound so the
// 16x column redundancy is free). EXEC stays all-ones through the WMMA loop.
__global__ void k_wmma_matvec(const float* __restrict__ W, const float* __restrict__ x,
                              const float* __restrict__ bias, const float* __restrict__ scalarAdd,
                              float* __restrict__ out, int rows, int K, int ld) {
    int lane = threadIdx.x;        // 0..31, wave32
    int tile = blockIdx.x;         // 16 rows per tile
    int half = lane >> 4;          // 0: K+0..1, 1: K+2..3
    int r16  = lane & 15;          // row within tile (A: lane = row)
    int row  = tile * 16 + r16;
    int rowc = row < rows ? row : rows - 1;   // branchless clamp (keeps EXEC full)
    const float* wrow = W + (size_t)rowc * ld;

    v8f c = {};
#pragma unroll 8
    for (int k = 0; k < K; k += 4) {
        int kOff = k + half * 2;
        v2f a, b;
        a.x = wrow[kOff];
        a.y = wrow[kOff + 1];
        b.x = x[kOff];          // broadcast vector into every B column
        b.y = x[kOff + 1];
        c = __builtin_amdgcn_wmma_f32_16x16x4_f32(false, a, false, b,
                                                  (short)0, c, false, false);
    }

    float sc = scalarAdd ? scalarAdd[0] : 0.0f;
    // D layout: lane 0 -> rows 0..7 (c[0..7]); lane 16 -> rows 8..15
    if (r16 == 0) {
        int mbase = tile * 16 + half * 8;
#pragma unroll
        for (int m = 0; m < 8; ++m) {
            int r = mbase + m;
            if (r < rows) out[r] = c[m] + (bias ? bias[r] : 0.0f) + sc;
        }
    }
}

// GRU gates + new hidden: r_out (ws, feeds attention & Wo) and hidden (d_out)
__global__ void k_gru(const float* __restrict__ gi, const float* __restrict__ gh,
                      const float* __restrict__ h, float* __restrict__ r_out,
                      float* __restrict__ hidden_out) {
    int i = blockIdx.x * blockDim.x + threadIdx.x;
    if (i < H) {
        float r = 1.0f / (1.0f + expf(-(gi[i] + gh[i])));
        float z = 1.0f / (1.0f + expf(-(gi[H + i] + gh[H + i])));
        float n = tanhf(gi[2 * H + i] + r * gh[2 * H + i]);
        float hv = h[i];
        float o = (1.0f - z) * n + z * hv;
        r_out[i] = o;
        hidden_out[i] = o;
    }
}

// v[j] = sum_i Wa[i][j] * r[i]   (transposed matvec, coalesced over j)
__global__ void k_tmatvec(const float* __restrict__ Wa, const float* __restrict__ r,
                          float* __restrict__ v) {
    int j = blockIdx.x * blockDim.x + threadIdx.x;
    float a0 = 0.f, a1 = 0.f, a2 = 0.f, a3 = 0.f;
    for (int i = 0; i < H; i += 4) {
        a0 += Wa[(size_t)(i + 0) * H + j] * r[i + 0];
        a1 += Wa[(size_t)(i + 1) * H + j] * r[i + 1];
        a2 += Wa[(size_t)(i + 2) * H + j] * r[i + 2];
        a3 += Wa[(size_t)(i + 3) * H + j] * r[i + 3];
    }
    v[j] = (a0 + a1) + (a2 + a3);
}

// scalar dot: out[0] = a . b   (ba . r term of the factored energies)
__global__ void k_dot(const float* __restrict__ a, const float* __restrict__ b,
                      int n, float* __restrict__ out) {
    __shared__ float red[256];
    int t = threadIdx.x;
    float s = 0.f;
    for (int i = t; i < n; i += 256) s += a[i] * b[i];
    red[t] = s;
    __syncthreads();
    for (int o = 128; o > 0; o >>= 1) {
        if (t < o) red[t] += red[t + o];
        __syncthreads();
    }
    if (t == 0) out[0] = red[0];
}

// log_softmax over n elements, single block (1024 threads), dual output
__global__ void k_logsoftmax(const float* __restrict__ in, float* __restrict__ out0,
                             float* __restrict__ out1, int n) {
    __shared__ float red[1024];
    int t = threadIdx.x;
    float m = -INFINITY;
    for (int i = t; i < n; i += 1024) m = fmaxf(m, in[i]);
    red[t] = m;
    __syncthreads();
    for (int o = 512; o > 0; o >>= 1) {
        if (t < o) red[t] = fmaxf(red[t], red[t + o]);
        __syncthreads();
    }
    float M = red[0];
    __syncthreads();
    float s = 0.f;
    for (int i = t; i < n; i += 1024) s += expf(in[i] - M);
    red[t] = s;
    __syncthreads();
    for (int o = 512; o > 0; o >>= 1) {
        if (t < o) red[t] += red[t + o];
        __syncthreads();
    }
    float L = M + logf(red[0]);
    for (int i = t; i < n; i += 1024) {
        float val = in[i] - L;
        out0[i] = val;
        if (out1) out1[i] = val;
    }
}

// context partials: partial[chunk][j] = sum_{i in chunk} attw[i]*enc[i][j]
__global__ void k_ctx_partial(const float* __restrict__ attw, const float* __restrict__ enc,
                              float* __restrict__ partial) {
    int j = blockIdx.x * blockDim.x + threadIdx.x;   // column
    int chunk = blockIdx.y;                          // 0..NCH-1
    int i0 = chunk * (S / NCH);
    float acc = 0.f;
    for (int i = i0; i < i0 + S / NCH; ++i)
        acc += attw[i] * enc[(size_t)i * H + j];
    partial[(size_t)chunk * H + j] = acc;
}

__global__ void k_ctx_reduce(const float* __restrict__ partial, float* __restrict__ ctx_out,
                             float* __restrict__ xo_ctx) {
    int j = blockIdx.x * blockDim.x + threadIdx.x;
    float acc = 0.f;
    for (int c = 0; c < NCH; ++c) acc += partial[(size_t)c * H + j];
    ctx_out[j] = acc;
    xo_ctx[j]  = acc;
}

extern "C" void kernel_launch(void* const* d_in, const int* in_sizes, int n_in,
                              void* d_out, int out_size, void* d_ws, size_t ws_size,
                              hipStream_t stream) {
    const int*   word      = (const int*)  d_in[0];
    const float* last_ctx  = (const float*)d_in[1];
    const float* last_hid  = (const float*)d_in[2];
    const float* enc       = (const float*)d_in[3];   // (S,1,H) contiguous = S x H
    const float* embedding = (const float*)d_in[4];
    const float* W_ih      = (const float*)d_in[5];
    const float* W_hh      = (const float*)d_in[6];
    const float* b_ih      = (const float*)d_in[7];
    const float* b_hh      = (const float*)d_in[8];
    const float* Wa        = (const float*)d_in[9];
    const float* ba        = (const float*)d_in[10];
    const float* Wo        = (const float*)d_in[11];
    const float* bo        = (const float*)d_in[12];

    float* ws   = (float*)d_ws;
    float* x    = ws;              // 2048  [emb | last_context]
    float* gi   = ws + 2048;       // 3072
    float* gh   = ws + 5120;       // 3072
    float* xo   = ws + 8192;       // 2048  [rnn_output | context]
    float* v    = ws + 10240;      // 1024  Wa^T r
    float* c0   = ws + 11264;      // 1     ba . r
    float* ener = ws + 11520;      // 4096
    float* attw = ws + 15616;      // 4096
    float* part = ws + 19712;      // NCH*1024
    float* oval = ws + 36096;      // 50257

    float* out      = (float*)d_out;
    float* o_output = out;                 // V
    float* o_ctx    = out + V;             // H
    float* o_hidden = out + V + H;         // H
    float* o_attn   = out + V + 2 * H;     // S

    // 1) x = [embedding[word], last_context]
    k_gather_x<<<8, 256, 0, stream>>>(word, embedding, last_ctx, x);

    // 2) gi = W_ih @ x + b_ih ; gh = W_hh @ h + b_hh   (WMMA GEMV)
    k_wmma_matvec<<<3 * H / 16, 32, 0, stream>>>(W_ih, x, b_ih, nullptr, gi, 3 * H, 2 * H, 2 * H);
    k_wmma_matvec<<<3 * H / 16, 32, 0, stream>>>(W_hh, last_hid, b_hh, nullptr, gh, 3 * H, H, H);

    // 3) GRU elementwise -> rnn_output (xo[0:H]) and hidden output
    k_gru<<<4, 256, 0, stream>>>(gi, gh, last_hid, xo, o_hidden);

    // 4) factored attention: energies = enc @ (Wa^T r) + (ba . r)
    k_tmatvec<<<4, 256, 0, stream>>>(Wa, xo, v);
    k_dot<<<1, 256, 0, stream>>>(ba, xo, H, c0);
    k_wmma_matvec<<<S / 16, 32, 0, stream>>>(enc, v, nullptr, c0, ener, S, H, H);

    // 5) attw = log_softmax(energies) -> ws + attention_weights output
    k_logsoftmax<<<1, 1024, 0, stream>>>(ener, attw, o_attn, S);

    // 6) context = attw @ enc -> context output and xo[H:2H]
    k_ctx_partial<<<dim3(4, NCH), 256, 0, stream>>>(attw, enc, part);
    k_ctx_reduce<<<4, 256, 0, stream>>>(part, o_ctx, xo + H);

    // 7) out_val = Wo @ [rnn_output|context] + bo   (412 MB stream — dominant)
    k_wmma_matvec<<<(V + 15) / 16, 32, 0, stream>>>(Wo, xo, bo, nullptr, oval, V, 2 * H, 2 * H);

    // 8) output = log_softmax(out_val)
    k_logsoftmax<<<1, 1024, 0, stream>>>(oval, o_output, nullptr, V);
}